// LegendreEncoder_14319420965216
// MI455X (gfx1250) — compile-verified
//
#include <hip/hip_runtime.h>

#define ORDER 128
#define LANES 32
#define WAVES_PER_BLOCK 4
#define ELEM_STRIDE_B 528                 /* 512B data + 16B pad: lane bank-quad stride = 4 -> conflict-free b128 */
#define TILE_B (LANES * ELEM_STRIDE_B)    /* 16896 B per tile */
#define NBUF 2

__global__ __launch_bounds__(WAVES_PER_BLOCK * 32)
void legendre_tdm_kernel(const float* __restrict__ x, float* __restrict__ out,
                         int ntiles, int wave_stride) {
    extern __shared__ __align__(16) char smem[];
    const int lane  = threadIdx.x & 31;
    const int wave  = threadIdx.x >> 5;
    const int gwave = blockIdx.x * WAVES_PER_BLOCK + wave;

    char* wave_smem = smem + (size_t)wave * (NBUF * TILE_B);

    int buf = 0;
    for (int tile = gwave; tile < ntiles; tile += wave_stride) {
        char* tbase = wave_smem + buf * TILE_B;

        // At most the previous iteration's 32 async stores are outstanding;
        // <=32 outstanding guarantees the stores sourced from *this* buffer
        // (two iterations ago) have drained, so we may overwrite it.
        asm volatile("s_wait_asynccnt 0x20" ::: "memory");

        const int   e  = tile * LANES + lane;   // ntiles*32 == nelem (exact)
        const float xv = x[e];

        float4* lrow = (float4*)(tbase + lane * ELEM_STRIDE_B);

        // P0..P3 into chunk 0
        float pm2 = 1.0f;   // P0
        float pm1 = xv;     // P1
        float4 c;
        c.x = pm2;
        c.y = pm1;
        {
            float p2 = 1.5f * xv * pm1 - 0.5f * pm2;                  // P2
            float p3 = (5.0f / 3.0f) * xv * p2 - (2.0f / 3.0f) * pm1; // P3
            c.z = p2; c.w = p3;
            pm2 = p2; pm1 = p3;
        }
        lrow[0] = c;  // ds_store_b128

        // Remaining chunks: n = 4c .. 4c+3, coefficients fold to literals.
        #pragma unroll
        for (int cidx = 1; cidx < ORDER / 4; ++cidx) {
            #pragma unroll
            for (int j = 0; j < 4; ++j) {
                const int   n = cidx * 4 + j;
                const float a = (2.0f * (float)n - 1.0f) / (float)n;
                const float b = ((float)n - 1.0f) / (float)n;
                const float pn = a * xv * pm1 - b * pm2;
                ((float*)&c)[j] = pn;
                pm2 = pm1; pm1 = pn;
            }
            lrow[cidx] = c;  // ds_store_b128, conflict-free (row stride 528B)
        }

        // LDS writes must be visible before the async engine reads them.
        asm volatile("s_wait_dscnt 0x0" ::: "memory");

        // Drain tile: instruction j writes element (tile*32+j)'s contiguous
        // 512B output block; lane l moves bytes [16l, 16l+16) of it straight
        // from LDS (no VGPR round trip), tracked by ASYNCcnt. Non-temporal:
        // 1 GB write-once stream must not thrash the 192 MB L2.
        const unsigned lds_base = (unsigned)(unsigned long long)tbase;
        const unsigned long long gbase =
            (unsigned long long)out + (unsigned long long)tile * (LANES * ORDER * 4ull);
        #pragma unroll
        for (int j = 0; j < LANES; ++j) {
            unsigned long long gaddr =
                gbase + (unsigned)(j * ORDER * 4) + (unsigned)(lane * 16);
            unsigned laddr = lds_base + (unsigned)(j * ELEM_STRIDE_B) + (unsigned)(lane * 16);
            asm volatile("global_store_async_from_lds_b128 %0, %1, off th:TH_STORE_NT"
                         :: "v"(gaddr), "v"(laddr)
                         : "memory");
        }

        buf ^= 1;
    }
    // s_endpgm performs an implicit wait-idle, covering outstanding ASYNCcnt.
}

extern "C" void kernel_launch(void* const* d_in, const int* in_sizes, int n_in,
                              void* d_out, int out_size, void* d_ws, size_t ws_size,
                              hipStream_t stream) {
    (void)n_in; (void)out_size; (void)d_ws; (void)ws_size;

    const float* x   = (const float*)d_in[0];
    float*       out = (float*)d_out;

    const int nelem  = in_sizes[0];        // 262144 * 8 = 2,097,152 (multiple of 32)
    const int ntiles = nelem / LANES;      // 65,536 tiles of 32 elements

    // ~8 tiles per wave keeps the async/double-buffer pipeline busy while
    // bounding grid size.
    const int tiles_per_wave = 8;
    int total_waves = (ntiles + tiles_per_wave - 1) / tiles_per_wave;
    int blocks      = (total_waves + WAVES_PER_BLOCK - 1) / WAVES_PER_BLOCK;
    if (blocks < 1) blocks = 1;
    const int wave_stride = blocks * WAVES_PER_BLOCK;

    const size_t shmem = (size_t)WAVES_PER_BLOCK * NBUF * TILE_B;  // 135,168 B

    hipLaunchKernelGGL(legendre_tdm_kernel,
                       dim3(blocks), dim3(WAVES_PER_BLOCK * 32), shmem, stream,
                       x, out, ntiles, wave_stride);
}